// Attention_91293824844342
// MI455X (gfx1250) — compile-verified
//
#include <hip/hip_runtime.h>
#include <hip/hip_bf16.h>

#define NEG_INF (-1.0e9f)
#define N_BATCH 64
#define T_LEN   2048
#define D_DIM   256
#define V_DIM   256

typedef __attribute__((ext_vector_type(2))) float v2f;
typedef __attribute__((ext_vector_type(8))) float v8f;

#if defined(__AMDGCN__) && __has_builtin(__builtin_amdgcn_wmma_f32_16x16x4_f32)
#define USE_WMMA_F32X4 1
#else
#define USE_WMMA_F32X4 0
#endif

// ---------------------------------------------------------------------------
// Kernel 1: energy[n,t] = dot(key[n,t,:], q[n,:]), masked to -1e9 for t>=len.
// Grid: N_BATCH * 8 blocks (8 segments of 256 t each), 256 threads (8 waves).
// Each wave computes 2 tiles of 16 t-rows via V_WMMA_F32_16X16X4_F32.
// ---------------------------------------------------------------------------
__global__ __launch_bounds__(256) void attn_energy(
    const float* __restrict__ query, const float* __restrict__ key,
    const int* __restrict__ lens, float* __restrict__ energy_out)
{
    __shared__ float qs[D_DIM];
    const int tid = threadIdx.x;
    const int n   = blockIdx.x >> 3;
    const int seg = blockIdx.x & 7;

    qs[tid] = query[n * D_DIM + tid];
    __syncthreads();

    const int len  = lens[n];
    const int wave = tid >> 5;
    const int lane = tid & 31;
    const int half = lane >> 4;   // 0: K=0,1 ; 1: K=2,3 (A/B layout halves)
    const int r    = lane & 15;   // A-matrix row M within tile

    const size_t keybase = (size_t)n * T_LEN * D_DIM;

    for (int tile = 0; tile < 2; ++tile) {
        const int t0 = seg * 256 + wave * 32 + tile * 16;
        const float* __restrict__ krow = key + keybase + (size_t)(t0 + r) * D_DIM;

#if USE_WMMA_F32X4
        v8f acc = {};
        #pragma unroll 4
        for (int d0 = 0; d0 < D_DIM; d0 += 4) {
            // A: 16x4 f32 tile of key (lane r holds K=2*half, 2*half+1)
            v2f a = *(const v2f*)(krow + d0 + 2 * half);
            // B: 4x16 f32, q chunk replicated across all 16 N columns
            v2f b;
            b.x = qs[d0 + 2 * half];
            b.y = qs[d0 + 2 * half + 1];
            acc = __builtin_amdgcn_wmma_f32_16x16x4_f32(
                false, a, false, b, (short)0, acc, false, false);
        }
        // D layout: VGPR i -> rows i (lanes 0-15) / i+8 (lanes 16-31); all N
        // columns identical, so lanes 0 and 16 each own 8 full energy rows.
        if (r == 0) {
            #pragma unroll
            for (int i = 0; i < 8; ++i) {
                const int t = t0 + half * 8 + i;
                float ev = acc[i];
                if (t >= len) ev = NEG_INF;
                energy_out[(size_t)n * T_LEN + t] = ev;
            }
        }
#else
        // Fallback: each lane-pair (r, r+16) splits the K dim mod 4.
        float p = 0.0f;
        #pragma unroll 4
        for (int d0 = 0; d0 < D_DIM; d0 += 4) {
            const float2 a = *(const float2*)(krow + d0 + 2 * half);
            const float2 b = *(const float2*)(qs + d0 + 2 * half);
            p += a.x * b.x + a.y * b.y;
        }
        p += __shfl_xor(p, 16, 32);
        if (half == 0) {
            const int t = t0 + r;
            energy_out[(size_t)n * T_LEN + t] = (t >= len) ? NEG_INF : p;
        }
#endif
    }
}

// ---------------------------------------------------------------------------
// Kernel 2: in-place softmax over T for each n. Grid: N_BATCH, 256 threads.
// ---------------------------------------------------------------------------
__global__ __launch_bounds__(256) void attn_softmax(float* __restrict__ attn)
{
    __shared__ float red[256];
    const int tid = threadIdx.x;
    float* __restrict__ a = attn + (size_t)blockIdx.x * T_LEN;

    float e[8];
    float m = -3.4e38f;
    #pragma unroll
    for (int i = 0; i < 8; ++i) {
        e[i] = a[tid + 256 * i];
        m = fmaxf(m, e[i]);
    }
    red[tid] = m;
    __syncthreads();
    for (int s = 128; s > 0; s >>= 1) {
        if (tid < s) red[tid] = fmaxf(red[tid], red[tid + s]);
        __syncthreads();
    }
    const float gmax = red[0];
    __syncthreads();

    float s = 0.0f;
    #pragma unroll
    for (int i = 0; i < 8; ++i) {
        e[i] = expf(e[i] - gmax);   // masked entries: exp(-1e9 - gmax) == 0
        s += e[i];
    }
    red[tid] = s;
    __syncthreads();
    for (int st = 128; st > 0; st >>= 1) {
        if (tid < st) red[tid] += red[tid + st];
        __syncthreads();
    }
    const float inv = 1.0f / red[0];

    #pragma unroll
    for (int i = 0; i < 8; ++i) a[tid + 256 * i] = e[i] * inv;
}

// ---------------------------------------------------------------------------
// Kernel 3: context[n,v] = sum_t attn[n,t] * value[n,t,v].
// Grid: N_BATCH * 4 blocks (64 v-columns each), 256 threads.
// Each thread owns a float4 of v and strides t by 16; value streamed b128.
// ---------------------------------------------------------------------------
__global__ __launch_bounds__(256) void attn_context(
    const float* __restrict__ attn, const float* __restrict__ value,
    float* __restrict__ ctx)
{
    __shared__ float  als[T_LEN];
    __shared__ float4 red[16][16];

    const int tid  = threadIdx.x;
    const int n    = blockIdx.x >> 2;
    const int vseg = blockIdx.x & 3;

    const float* __restrict__ a = attn + (size_t)n * T_LEN;
    #pragma unroll
    for (int i = 0; i < 8; ++i) als[tid + 256 * i] = a[tid + 256 * i];
    __syncthreads();

    const int tv = tid & 15;   // which float4 of the 64-wide v tile
    const int tg = tid >> 4;   // t-group 0..15
    const int v4 = vseg * 64 + tv * 4;
    const float* __restrict__ vb = value + (size_t)n * T_LEN * V_DIM + v4;

    float4 acc = make_float4(0.f, 0.f, 0.f, 0.f);
    #pragma unroll 4
    for (int t = tg; t < T_LEN; t += 16) {
        const float  av  = als[t];
        const float4 val = *(const float4*)(vb + (size_t)t * V_DIM);
        acc.x += av * val.x;
        acc.y += av * val.y;
        acc.z += av * val.z;
        acc.w += av * val.w;
    }
    red[tg][tv] = acc;
    __syncthreads();

    if (tg == 0) {
        float4 c = red[0][tv];
        #pragma unroll
        for (int g = 1; g < 16; ++g) {
            const float4 p = red[g][tv];
            c.x += p.x; c.y += p.y; c.z += p.z; c.w += p.w;
        }
        *(float4*)(ctx + (size_t)n * V_DIM + v4) = c;
    }
}

extern "C" void kernel_launch(void* const* d_in, const int* in_sizes, int n_in,
                              void* d_out, int out_size, void* d_ws, size_t ws_size,
                              hipStream_t stream) {
    const float* query = (const float*)d_in[0];   // (64, 256)
    const float* key   = (const float*)d_in[1];   // (64, 2048, 256)
    const float* value = (const float*)d_in[2];   // (64, 2048, 256)
    const int*   lens  = (const int*)d_in[3];     // (64,)

    float* out  = (float*)d_out;
    float* ctx  = out;                            // (64, 256) first
    float* attn = out + N_BATCH * V_DIM;          // (64, 2048) second

    attn_energy <<<N_BATCH * 8, 256, 0, stream>>>(query, key, lens, attn);
    attn_softmax<<<N_BATCH,     256, 0, stream>>>(attn);
    attn_context<<<N_BATCH * 4, 256, 0, stream>>>(attn, value, ctx);
}